// OrthogonalLayer_21560735826048
// MI455X (gfx1250) — compile-verified
//
#include <hip/hip_runtime.h>

typedef __attribute__((ext_vector_type(2))) float v2f;
typedef __attribute__((ext_vector_type(8))) float v8f;

// D = A(16x4) x B(4x16) + C, fp32 WMMA (CDNA5, wave32)
#define WMMA_F32X4(a, b, c) \
    __builtin_amdgcn_wmma_f32_16x16x4_f32(false, (a), false, (b), (short)0, (c), false, false)

#define NUM_ITERS 5
#define NS_ITERS 10   // 1 analytic first step (Z0=I) + 9 general coupled steps
#define MQ 256
#define RR 16
#define QS 17   // LDS row stride (coprime with 64 banks -> conflict-free)

// Complex 16x16 matmul D = X * Y executed by ONE full wave (EXEC all ones).
// All matrices row-major in LDS with row stride QS.
__device__ __forceinline__ void cmul16(float (*Dr)[QS], float (*Di)[QS],
                                       float (*Xr)[QS], float (*Xi)[QS],
                                       float (*Yr)[QS], float (*Yi)[QS],
                                       int lane) {
    const int lr = lane & 15;
    const int lh = lane >> 4;
    v8f crr = {}, cii = {}, cri = {}, cir = {};
#pragma unroll
    for (int kk = 0; kk < 16; kk += 4) {
        const int k = kk + 2 * lh;
        v2f ar, ai, br, bi;
        ar.x = Xr[lr][k];     ar.y = Xr[lr][k + 1];
        ai.x = Xi[lr][k];     ai.y = Xi[lr][k + 1];
        br.x = Yr[k][lr];     br.y = Yr[k + 1][lr];
        bi.x = Yi[k][lr];     bi.y = Yi[k + 1][lr];
        crr = WMMA_F32X4(ar, br, crr);
        cii = WMMA_F32X4(ai, bi, cii);
        cri = WMMA_F32X4(ar, bi, cri);
        cir = WMMA_F32X4(ai, br, cir);
    }
#pragma unroll
    for (int v = 0; v < 8; ++v) {
        Dr[8 * lh + v][lr] = crr[v] - cii[v];
        Di[8 * lh + v][lr] = cri[v] + cir[v];
    }
}

__global__ __launch_bounds__(256, 1)
void ortho_wmma_kernel(const float* __restrict__ x, float* __restrict__ out) {
    __shared__ float sQr[MQ][QS], sQi[MQ][QS];      // resident Q (re/im)
    __shared__ float sPart[8][16][16];              // per-wave gram partials
    __shared__ float sGr[16][QS], sGi[16][QS];      // gram, later W = gram^{-1/2}
    __shared__ float sY0r[16][QS], sY0i[16][QS], sY1r[16][QS], sY1i[16][QS];
    __shared__ float sZ0r[16][QS], sZ0i[16][QS], sZ1r[16][QS], sZ1i[16][QS];
    __shared__ float sTr[16][QS], sTi[16][QS];
    __shared__ float sScale;

    const int tid  = threadIdx.x;
    const int lane = tid & 31;
    const int wave = tid >> 5;
    const int lr   = lane & 15;
    const int lh   = lane >> 4;
    const int ei   = tid >> 4;      // element row for 256-thread elementwise ops
    const int ej   = tid & 15;      // element col

    const float* xb = x + (size_t)blockIdx.x * (MQ * RR * 2);
    float*       ob = out + (size_t)blockIdx.x * (MQ * RR * 2);

    // ---- load Q (interleaved re/im) into LDS, fully coalesced float4 ----
    const float4* xb4 = (const float4*)xb;
    for (int f = tid; f < (MQ * RR * 2) / 4; f += 256) {
        float4 d = xb4[f];
        int m = f >> 3;
        int r0 = (f & 7) * 2;
        sQr[m][r0]     = d.x;  sQi[m][r0]     = d.y;
        sQr[m][r0 + 1] = d.z;  sQi[m][r0 + 1] = d.w;
    }
    __syncthreads();

#pragma unroll 1
    for (int it = 0; it < NUM_ITERS; ++it) {
        // ================= gram = Q^H Q  (per-wave 32-row slabs) =================
        v8f cRR = {}, cII = {}, cRI = {}, cIR = {};
        const int m0 = wave * 32;
#pragma unroll
        for (int kk = 0; kk < 32; kk += 4) {
            const int m = m0 + kk + 2 * lh;
            v2f qr, qi;
            qr.x = sQr[m][lr];      qr.y = sQr[m + 1][lr];
            qi.x = sQi[m][lr];      qi.y = sQi[m + 1][lr];
            // A = Qx^T and B = Qy read identical lane data -> reuse registers
            cRR = WMMA_F32X4(qr, qr, cRR);
            cII = WMMA_F32X4(qi, qi, cII);
            cRI = WMMA_F32X4(qr, qi, cRI);
            cIR = WMMA_F32X4(qi, qr, cIR);
        }
        // reduce real part across 8 waves
#pragma unroll
        for (int v = 0; v < 8; ++v)
            sPart[wave][8 * lh + v][lr] = cRR[v] + cII[v];
        __syncthreads();
        {
            float s = 0.f;
#pragma unroll
            for (int w = 0; w < 8; ++w) s += sPart[w][ei][ej];
            sGr[ei][ej] = s;
        }
        __syncthreads();
        // reduce imaginary part (reuse staging buffer)
#pragma unroll
        for (int v = 0; v < 8; ++v)
            sPart[wave][8 * lh + v][lr] = cRI[v] - cIR[v];
        __syncthreads();
        {
            float s = 0.f;
#pragma unroll
            for (int w = 0; w < 8; ++w) s += sPart[w][ei][ej];
            sGi[ei][ej] = s;
        }
        __syncthreads();

        // ================= scale: Y = G * (16/trace) =================
        if (tid == 0) {
            float t = 0.f;
#pragma unroll
            for (int i = 0; i < 16; ++i) t += sGr[i][i];
            t = fmaxf(t, 1e-20f);
            sScale = 16.0f / t;
        }
        __syncthreads();
        {
            const float ic = sScale;
            sY0r[ei][ej] = sGr[ei][ej] * ic;
            sY0i[ei][ej] = sGi[ei][ej] * ic;
        }
        __syncthreads();

        // ===== Newton-Schulz: G^{-1/2} (replaces eigh; SPD inverse-sqrt unique) ====
        // Step 0 (Z0 = I): T0 = 1.5I - 0.5*Y0;  Z1 = T0 (copy);  Y1 = Y0*T0.
        {
            float tr = ((ei == ej) ? 1.5f : 0.0f) - 0.5f * sY0r[ei][ej];
            float ti = -0.5f * sY0i[ei][ej];
            sTr[ei][ej]  = tr;  sTi[ei][ej]  = ti;
            sZ1r[ei][ej] = tr;  sZ1i[ei][ej] = ti;
        }
        __syncthreads();
        if (wave == 0) cmul16(sY1r, sY1i, sY0r, sY0i, sTr, sTi, lane);
        __syncthreads();

        // General coupled steps; ping-pong pointers tracked uniformly by all threads
        float (*Yr)[QS] = sY1r, (*Yi)[QS] = sY1i;
        float (*Zr)[QS] = sZ1r, (*Zi)[QS] = sZ1i;
        float (*Ynr)[QS] = sY0r, (*Yni)[QS] = sY0i;
        float (*Znr)[QS] = sZ0r, (*Zni)[QS] = sZ0i;
#pragma unroll 1
        for (int ns = 1; ns < NS_ITERS; ++ns) {
            if (wave == 0) cmul16(sTr, sTi, Zr, Zi, Yr, Yi, lane);   // P = Z*Y
            __syncthreads();
            // T = 1.5I - 0.5P  (all 256 threads, 1 element each)
            sTr[ei][ej] = ((ei == ej) ? 1.5f : 0.0f) - 0.5f * sTr[ei][ej];
            sTi[ei][ej] = -0.5f * sTi[ei][ej];
            __syncthreads();
            if (wave == 0)      cmul16(Ynr, Yni, Yr, Yi, sTr, sTi, lane); // Y <- Y*T
            else if (wave == 1) cmul16(Znr, Zni, sTr, sTi, Zr, Zi, lane); // Z <- T*Z
            __syncthreads();
            float (*t0)[QS];
            t0 = Yr; Yr = Ynr; Ynr = t0;   t0 = Yi; Yi = Yni; Yni = t0;
            t0 = Zr; Zr = Znr; Znr = t0;   t0 = Zi; Zi = Zni; Zni = t0;
        }
        // W = Z / sqrt(c) = Z * sqrt(sScale)  -> store into sGr/sGi
        {
            const float fs = sqrtf(sScale);
            sGr[ei][ej] = Zr[ei][ej] * fs;
            sGi[ei][ej] = Zi[ei][ej] * fs;
        }
        __syncthreads();

        // ================= Q <- Q * W  (two 16-row tiles per wave) =================
#pragma unroll
        for (int t = 0; t < 2; ++t) {
            const int mt = wave * 32 + t * 16;
            v8f drr = {}, dii = {}, dri = {}, dir = {};
#pragma unroll
            for (int kk = 0; kk < 16; kk += 4) {
                const int k = kk + 2 * lh;
                v2f ar, ai, br, bi;
                ar.x = sQr[mt + lr][k];   ar.y = sQr[mt + lr][k + 1];
                ai.x = sQi[mt + lr][k];   ai.y = sQi[mt + lr][k + 1];
                br.x = sGr[k][lr];        br.y = sGr[k + 1][lr];
                bi.x = sGi[k][lr];        bi.y = sGi[k + 1][lr];
                drr = WMMA_F32X4(ar, br, drr);
                dii = WMMA_F32X4(ai, bi, dii);
                dri = WMMA_F32X4(ar, bi, dri);
                dir = WMMA_F32X4(ai, br, dir);
            }
#pragma unroll
            for (int v = 0; v < 8; ++v) {
                sQr[mt + 8 * lh + v][lr] = drr[v] - dii[v];
                sQi[mt + 8 * lh + v][lr] = dri[v] + dir[v];
            }
        }
        __syncthreads();
    }

    // ---- store result (re-interleave), coalesced float4 ----
    float4* ob4 = (float4*)ob;
    for (int f = tid; f < (MQ * RR * 2) / 4; f += 256) {
        int m = f >> 3;
        int r0 = (f & 7) * 2;
        float4 d;
        d.x = sQr[m][r0];      d.y = sQi[m][r0];
        d.z = sQr[m][r0 + 1];  d.w = sQi[m][r0 + 1];
        ob4[f] = d;
    }
}

extern "C" void kernel_launch(void* const* d_in, const int* in_sizes, int n_in,
                              void* d_out, int out_size, void* d_ws, size_t ws_size,
                              hipStream_t stream) {
    const float* x = (const float*)d_in[0];
    float* out = (float*)d_out;
    const int batches = in_sizes[0] / (MQ * RR * 2);   // 4096
    ortho_wmma_kernel<<<batches, 256, 0, stream>>>(x, out);
}